// MyGNN_10548439679440
// MI455X (gfx1250) — compile-verified
//
#include <hip/hip_runtime.h>

typedef __attribute__((ext_vector_type(2))) float v2f;
typedef __attribute__((ext_vector_type(8))) float v8f;

#define GIN_N     50000
#define GIN_E     800000
#define GIN_FIN   64
#define GIN_FHID  128
#define GIN_FOUT  2
#define GIN_MT    (GIN_N / 16)     // 3125 exact M-tiles
#define GIN_EPS1  1.0f             // (1 + eps), eps = 0 in reference

// ---------------------------------------------------------------- zero fill
__global__ __launch_bounds__(256)
void gin_zero_f4(float4* __restrict__ p, int n4) {
  int i = blockIdx.x * 256 + threadIdx.x;
  if (i < n4) p[i] = make_float4(0.f, 0.f, 0.f, 0.f);
}

// ------------------------------------------------- edge scatter, 64 features
// thread t -> edge e = t>>6, feature f = t&63. Feature in the low bits =>
// each wave32 touches 32 consecutive floats of one destination row
// (coalesced global_atomic_add_f32); src/dst loads are wave-uniform.
__global__ __launch_bounds__(256)
void gin_scatter64(const float* __restrict__ feat, const int* __restrict__ src,
                   const int* __restrict__ dst, float* __restrict__ agg) {
  int t = blockIdx.x * 256 + threadIdx.x;
  int e = t >> 6;
  if (e >= GIN_E) return;
  int f = t & 63;
  int s = src[e], d = dst[e];
  atomicAdd(agg + d * GIN_FIN + f, feat[s * GIN_FIN + f]);
}

// ------------------------------------------------ edge scatter, 128 features
__global__ __launch_bounds__(256)
void gin_scatter128(const float* __restrict__ feat, const int* __restrict__ src,
                    const int* __restrict__ dst, float* __restrict__ agg) {
  int t = blockIdx.x * 256 + threadIdx.x;
  int e = t >> 7;
  if (e >= GIN_E) return;
  int f = t & 127;
  int s = src[e], d = dst[e];
  atomicAdd(agg + d * GIN_FHID + f, feat[s * GIN_FHID + f]);
}

// ----------------------- layer 1: h = relu(((1+eps)x + agg1) @ W1^T + b1)
// Block = one 16-row M-tile. A tile (16x64 fp32) staged in LDS with row
// stride 68 (conflict-free column reads). 8 waves -> 8 N-tiles of 16.
// K = 64 -> 16 x V_WMMA_F32_16X16X4_F32 per wave.
__global__ __launch_bounds__(256)
void gin_gemm1(const float* __restrict__ x, const float* __restrict__ agg1,
               const float* __restrict__ W1, const float* __restrict__ b1,
               float* __restrict__ h) {
  __shared__ float As[16][68];

  const int mtile = blockIdx.x;          // 0..3124
  const int tid   = threadIdx.x;         // 0..255

  // Stage A = (1+eps)*x + agg1 for this 16x64 tile (4 elems/thread).
  for (int i = tid; i < 16 * GIN_FIN; i += 256) {
    int m = i >> 6, k = i & 63;
    int g = (mtile * 16 + m) * GIN_FIN + k;
    As[m][k] = GIN_EPS1 * x[g] + agg1[g];
  }
  __syncthreads();

  const int wave = tid >> 5;             // N-tile 0..7
  const int lane = tid & 31;
  const int nloc = lane & 15;
  const int half = lane >> 4;            // 0: K offset 0/1, 1: K offset 2/3
  const int mrow = lane & 15;
  const int n    = wave * 16 + nloc;

  v8f c = {};
  #pragma unroll
  for (int kt = 0; kt < GIN_FIN; kt += 4) {
    const int k0 = kt + 2 * half;
    v2f a, b;
    a.x = As[mrow][k0];                  // A 16x4: VGPR0 K = {0,2}
    a.y = As[mrow][k0 + 1];              //         VGPR1 K = {1,3}
    b.x = W1[n * GIN_FIN + k0];          // B[k][n] = W1[n][k]
    b.y = W1[n * GIN_FIN + k0 + 1];
    c = __builtin_amdgcn_wmma_f32_16x16x4_f32(
        /*neg_a=*/false, a, /*neg_b=*/false, b,
        /*c_mod=*/(short)0, c, /*reuse_a=*/false, /*reuse_b=*/false);
  }

  const float bias = b1[n];
  #pragma unroll
  for (int v = 0; v < 8; ++v) {          // D: VGPR v -> row v + 8*half
    int m = mtile * 16 + v + 8 * half;
    float val = c[v] + bias;
    h[m * GIN_FHID + n] = val > 0.f ? val : 0.f;
  }
}

// --------------------------- layer 2: out = ((h + agg2) @ W2^T + b2), N=2
// One wave per 16-row M-tile; N padded to 16 with zero B columns.
// K = 128 -> 32 x V_WMMA_F32_16X16X4_F32. Guard is wave-uniform so EXEC
// stays all-1s through every WMMA.
__global__ __launch_bounds__(256)
void gin_gemm2(const float* __restrict__ h, const float* __restrict__ agg2,
               const float* __restrict__ W2, const float* __restrict__ b2,
               float* __restrict__ out) {
  const int gwave = (blockIdx.x * 256 + threadIdx.x) >> 5;
  if (gwave >= GIN_MT) return;           // wave-uniform exit

  const int lane = threadIdx.x & 31;
  const int nloc = lane & 15;
  const int half = lane >> 4;
  const int mrow = lane & 15;
  const int abase = (gwave * 16 + mrow) * GIN_FHID;

  v8f c = {};
  #pragma unroll
  for (int kt = 0; kt < GIN_FHID; kt += 4) {
    const int k0 = kt + 2 * half;
    v2f a, b;
    a.x = h[abase + k0]     + agg2[abase + k0];
    a.y = h[abase + k0 + 1] + agg2[abase + k0 + 1];
    b.x = (nloc < GIN_FOUT) ? W2[nloc * GIN_FHID + k0]     : 0.f;
    b.y = (nloc < GIN_FOUT) ? W2[nloc * GIN_FHID + k0 + 1] : 0.f;
    c = __builtin_amdgcn_wmma_f32_16x16x4_f32(
        false, a, false, b, (short)0, c, false, false);
  }

  if (nloc < GIN_FOUT) {
    const float bias = b2[nloc];
    #pragma unroll
    for (int v = 0; v < 8; ++v) {
      int m = gwave * 16 + v + 8 * half;
      out[m * GIN_FOUT + nloc] = c[v] + bias;
    }
  }
}

// --------------------------------------------------------------------------
extern "C" void kernel_launch(void* const* d_in, const int* in_sizes, int n_in,
                              void* d_out, int out_size, void* d_ws, size_t ws_size,
                              hipStream_t stream) {
  const float* x  = (const float*)d_in[0];
  const int*   ei = (const int*)  d_in[1];   // [2, E]: row0 = src, row1 = dst
  const float* W1 = (const float*)d_in[2];   // [128, 64]
  const float* b1 = (const float*)d_in[3];
  const float* W2 = (const float*)d_in[4];   // [2, 128]
  const float* b2 = (const float*)d_in[5];
  float* out = (float*)d_out;                // [50000, 2]

  const int* src = ei;
  const int* dst = ei + GIN_E;

  // Workspace: agg1 [12.8MB] | h [25.6MB] | agg2 [25.6MB]  (64MB total)
  float* ws   = (float*)d_ws;
  float* agg1 = ws;
  float* h    = ws + (size_t)GIN_N * GIN_FIN;
  float* agg2 = ws + (size_t)GIN_N * (GIN_FIN + GIN_FHID);

  // Zero the accumulation buffers every call (deterministic, capture-safe).
  {
    int n4 = GIN_N * GIN_FIN / 4;
    gin_zero_f4<<<(n4 + 255) / 256, 256, 0, stream>>>((float4*)agg1, n4);
  }
  {
    int n4 = GIN_N * GIN_FHID / 4;
    gin_zero_f4<<<(n4 + 255) / 256, 256, 0, stream>>>((float4*)agg2, n4);
  }

  // Layer 1: scatter + fused (add, GEMM fp32-WMMA, bias, ReLU)
  {
    int nt = GIN_E * GIN_FIN;  // 51.2M threads
    gin_scatter64<<<(nt + 255) / 256, 256, 0, stream>>>(x, src, dst, agg1);
  }
  gin_gemm1<<<GIN_MT, 256, 0, stream>>>(x, agg1, W1, b1, h);

  // Layer 2: scatter + fused (add, GEMM fp32-WMMA, bias)
  {
    int nt = GIN_E * GIN_FHID; // 102.4M threads
    gin_scatter128<<<(nt + 255) / 256, 256, 0, stream>>>(h, src, dst, agg2);
  }
  gin_gemm2<<<(GIN_MT + 7) / 8, 256, 0, stream>>>(h, agg2, W2, b2, out);
}